// GraphLearner_4793183502935
// MI455X (gfx1250) — compile-verified
//
#include <hip/hip_runtime.h>
#include <math.h>

typedef __attribute__((ext_vector_type(2))) float v2f;
typedef __attribute__((ext_vector_type(8))) float v8f;

#define DD 512
#define CNUM 4

// ws layout:
//   V : float[2][16][512]  (zero-padded A matrices, 64 KB)
//   G : float[2][16][512]  (GEMM outputs, 64 KB)

// -------------------------------------------------------------------------
// Kernel 1: per (gcn, batch) wave -> cosine gram (edge embedding), GCN
// normalization, fold norm_adj row 0 into the raw node features:
//   V[g][b][:] = sum_j dinv0 * adj[0][j] * dinvj * node_j[:]
// node_0 = bt[b]; node_1..4 = (g==0 ? bt : bi)[0..3]
// Each wave also zeroes 3 of the 12 padding rows (rows 4..15) of its V[g].
// -------------------------------------------------------------------------
__global__ __launch_bounds__(256) void build_v_kernel(
    const float* __restrict__ bt, const float* __restrict__ bi,
    float* __restrict__ V)
{
  const int wave = threadIdx.x >> 5;   // 0..7
  const int lane = threadIdx.x & 31;
  const int g = wave >> 2;             // 0 = tt, 1 = it
  const int b = wave & 3;
  const float* nodes = (g == 0) ? bt : bi;

  // 5 node vectors, 16 elements per lane (stride 32)
  float x[5][16];
  for (int e = 0; e < 16; ++e) {
    const int idx = lane + e * 32;
    x[0][e] = bt[b * DD + idx];
    for (int j = 1; j < 5; ++j) x[j][e] = nodes[(j - 1) * DD + idx];
  }

  // all pairwise dots (symmetric), wave32 butterfly reduction
  float dot[5][5];
  for (int i = 0; i < 5; ++i)
    for (int j = i; j < 5; ++j) {
      float p = 0.f;
      for (int e = 0; e < 16; ++e) p += x[i][e] * x[j][e];
      for (int off = 16; off > 0; off >>= 1) p += __shfl_xor(p, off, 32);
      dot[i][j] = p; dot[j][i] = p;
    }

  float inrm[5];
  for (int i = 0; i < 5; ++i)
    inrm[i] = 1.0f / fmaxf(sqrtf(dot[i][i]), 1e-12f);

  float adj[5][5], rsum[5];
  for (int i = 0; i < 5; ++i) {
    float s = 0.f;
    for (int j = 0; j < 5; ++j) {
      const float c = dot[i][j] * inrm[i] * inrm[j];
      const float a = fmaxf(c, 0.f) + (i == j ? 1.0f : 0.0f);
      adj[i][j] = a; s += a;
    }
    rsum[i] = s;
  }
  float dinv[5];
  for (int i = 0; i < 5; ++i) dinv[i] = (rsum[i] > 0.f) ? rsqrtf(rsum[i]) : 0.f;

  float w[5];
  for (int j = 0; j < 5; ++j) w[j] = dinv[0] * adj[0][j] * dinv[j];

  float* Vg = V + g * 16 * DD;
  for (int e = 0; e < 16; ++e) {
    float acc = 0.f;
    for (int j = 0; j < 5; ++j) acc += w[j] * x[j][e];
    Vg[b * DD + lane + e * 32] = acc;
  }

  // zero the padding rows: wave (g,b) clears rows 4+3b .. 6+3b of V[g]
  for (int rr = 0; rr < 3; ++rr) {
    float* row = Vg + (4 + b * 3 + rr) * DD;
    for (int e = 0; e < 16; ++e) row[lane + e * 32] = 0.0f;
  }
}

// -------------------------------------------------------------------------
// Kernel 2: G[g] (16x512) = V[g] (16x512, rows 4..15 zero) @ W[g] (512x512)
// One wave per (g, 16-wide N tile); 128 steps of v_wmma_f32_16x16x4_f32.
// A fragment (16x4 f32): VGPR0 = {K0 lanes 0-15, K2 lanes 16-31},
//                        VGPR1 = {K1, K3}. B fragment mirrors (KxN rows).
// C/D (16x16 f32): lane l, vgpr v -> D[v + 8*(l>>4)][l&15].
// No divergence anywhere -> EXEC stays all-1s as WMMA requires.
// -------------------------------------------------------------------------
__global__ __launch_bounds__(128) void wmma_gemm_kernel(
    const float* __restrict__ W_tt, const float* __restrict__ W_it,
    const float* __restrict__ V, float* __restrict__ G)
{
  const int wid  = blockIdx.x * 4 + (threadIdx.x >> 5); // 0..63
  const int lane = threadIdx.x & 31;
  const int g     = wid >> 5;    // 0..1
  const int ntile = wid & 31;    // 0..31
  const float* Wm = g ? W_it : W_tt;
  const float* Vm = V + g * 16 * DD;

  const int r    = lane & 15;
  const int half = lane >> 4;
  const int n    = ntile * 16 + r;

  const float* aptr = Vm + r * DD + 2 * half;          // +k0 per step
  const float* bptr = Wm + (2 * half) * DD + n;        // +k0*DD per step

  v8f c = {};
  for (int k0 = 0; k0 < DD; k0 += 4) {
    const v2f a = *(const v2f*)(aptr + k0);             // b64 load
    v2f bf;
    bf.x = bptr[(size_t)k0 * DD];
    bf.y = bptr[(size_t)(k0 + 1) * DD];
    c = __builtin_amdgcn_wmma_f32_16x16x4_f32(
        /*neg_a=*/false, a, /*neg_b=*/false, bf,
        /*c_mod=*/(short)0, c, /*reuse_a=*/false, /*reuse_b=*/false);
  }

  float* Gt = G + g * 16 * DD + ntile * 16;
  for (int v = 0; v < 8; ++v) {
    const int m = v + 8 * half;
    Gt[m * DD + r] = c[v];
  }
}

// -------------------------------------------------------------------------
// Kernel 3: refined = 0.5*bt + 0.5*(0.7*tanh(Gtt+btt) + 0.3*tanh(Git+bit))
// -------------------------------------------------------------------------
__global__ __launch_bounds__(256) void epilogue_kernel(
    const float* __restrict__ bt, const float* __restrict__ btt,
    const float* __restrict__ bit, const float* __restrict__ G,
    float* __restrict__ out)
{
  const int idx = blockIdx.x * blockDim.x + threadIdx.x; // 0..2047
  if (idx >= CNUM * DD) return;
  const int b = idx >> 9;
  const int e = idx & (DD - 1);
  const float gtt = tanhf(G[0 * 16 * DD + b * DD + e] + btt[idx]);
  const float git = tanhf(G[1 * 16 * DD + b * DD + e] + bit[idx]);
  const float graph = 0.7f * gtt + 0.3f * git;
  out[idx] = 0.5f * bt[idx] + 0.5f * graph;
}

extern "C" void kernel_launch(void* const* d_in, const int* in_sizes, int n_in,
                              void* d_out, int out_size, void* d_ws, size_t ws_size,
                              hipStream_t stream) {
  const float* bt   = (const float*)d_in[0];
  const float* bi   = (const float*)d_in[1];
  const float* img  = (const float*)d_in[2];
  const float* W_tt = (const float*)d_in[3];
  const float* b_tt = (const float*)d_in[4];
  const float* W_it = (const float*)d_in[5];
  const float* b_it = (const float*)d_in[6];
  float* out = (float*)d_out;

  float* V = (float*)d_ws;        // 2*16*512 floats (64 KB), zero-padded A
  float* G = V + 2 * 16 * DD;     // 2*16*512 floats (64 KB)

  // Bandwidth-dominant part: 80 MB passthrough at HBM peak.
  hipMemcpyAsync(out + CNUM * DD, img,
                 (size_t)in_sizes[2] * sizeof(float),
                 hipMemcpyDeviceToDevice, stream);

  build_v_kernel<<<1, 256, 0, stream>>>(bt, bi, V);
  wmma_gemm_kernel<<<16, 128, 0, stream>>>(W_tt, W_it, V, G);
  epilogue_kernel<<<(CNUM * DD + 255) / 256, 256, 0, stream>>>(bt, b_tt, b_it, G, out);
}